// DirectionalAttn_18966575579527
// MI455X (gfx1250) — compile-verified
//
#include <hip/hip_runtime.h>
#include <math.h>

// ---------------------------------------------------------------------------
// DirectionalAttn for MI455X (gfx1250, wave32, WMMA).
// B=16, S=128, D=256, fp32. All GEMMs use V_WMMA_F32_16X16X4_F32.
// Each wave owns a 16x32 output strip: one A fragment feeds two back-to-back
// WMMAs (independent accumulator chains c0/c1), halving A-side VMEM per FLOP.
// ---------------------------------------------------------------------------

typedef __attribute__((ext_vector_type(2))) float v2f;
typedef __attribute__((ext_vector_type(8))) float v8f;

constexpr float ATT_SCALE = 5.0f;

enum EpiMode { EPI_NONE = 0, EPI_BIAS = 1, EPI_ELU_BIAS = 2 };

__device__ __forceinline__ float elu_act(float v) {
    return (v > 0.0f) ? v : (__expf(v) - 1.0f);
}

// ---------------------------------------------------------------------------
// fp32 WMMA GEMM: C[M,N] = epi(A[M,K] * W[K,N] + bias[N])
// Block = 128 threads = 4 waves; wave w handles cols [(bx*4+w)*32, +32).
// Grid = (N/128, M/16).
//
// Fragment layout (CDNA5 ISA 7.12.2, wave32):
//   A (16x4 f32, 2 VGPRs): lane l<16 -> A[m0+l][k+0..1], lane l>=16 -> k+2..3
//   B (4x16 f32, 2 VGPRs): lane l<16 -> W[k+0..1][n0+l], lane l>=16 -> k+2..3
//   C/D (16x16 f32, 8 VGPRs): vgpr r, half h, lane lm -> row m0+r+8h, col n0+lm
// ---------------------------------------------------------------------------
template <int EPI>
__global__ __launch_bounds__(128)
void gemm16x32_wmma(const float* __restrict__ A, const float* __restrict__ W,
                    const float* __restrict__ bias, float* __restrict__ C,
                    int M, int K, int N)
{
    const int lane = threadIdx.x & 31;
    const int wave = threadIdx.x >> 5;
    const int lm   = lane & 15;
    const int half = lane >> 4;

    const int m0 = blockIdx.y * 16;
    const int n0 = (blockIdx.x * 4 + wave) * 32;

    const float* __restrict__ Arow = A + (size_t)(m0 + lm) * K + 2 * half;
    const float* __restrict__ W0   = W + n0 + lm;
    const float* __restrict__ W1   = W + n0 + 16 + lm;

    v8f c0 = {}, c1 = {};
#pragma unroll 4
    for (int k = 0; k < K; k += 4) {
        const int kk = k + 2 * half;
        v2f a = *(const v2f*)(Arow + k);               // 8B-aligned b64 load
        v2f b0, b1;
        b0.x = W0[(size_t)kk * N];
        b0.y = W0[(size_t)(kk + 1) * N];
        b1.x = W1[(size_t)kk * N];
        b1.y = W1[(size_t)(kk + 1) * N];
        // Same A operand reused across two WMMAs; independent C chains.
        c0 = __builtin_amdgcn_wmma_f32_16x16x4_f32(false, a, false, b0,
                                                   (short)0, c0, false, false);
        c1 = __builtin_amdgcn_wmma_f32_16x16x4_f32(false, a, false, b1,
                                                   (short)0, c1, false, false);
    }

    const int col0 = n0 + lm;
    const int col1 = n0 + 16 + lm;
    float bv0 = 0.0f, bv1 = 0.0f;
    if (EPI != EPI_NONE) { bv0 = bias[col0]; bv1 = bias[col1]; }

#pragma unroll
    for (int r = 0; r < 8; ++r) {
        const int row = m0 + 8 * half + r;
        float v0 = c0[r] + bv0;
        float v1 = c1[r] + bv1;
        if (EPI == EPI_ELU_BIAS) { v0 = elu_act(v0); v1 = elu_act(v1); }
        C[(size_t)row * N + col0] = v0;
        C[(size_t)row * N + col1] = v1;
    }
}

// ---------------------------------------------------------------------------
// Fused directional attention. One block per (b, i); one thread per channel d.
// logits are bounded in [-SCALE, SCALE] by tanh, so the per-channel softmax
// needs no running max: num = sum_j exp(l_j)*rep[j,d], den = sum_j exp(l_j).
// `dep` already contains attn_bias (folded into its GEMM epilogue).
// Masked / future positions contribute exp(-1e30) == 0 in the reference; the
// i==0 (den==0) row collapses to 0 exactly as attn_score*mask does.
// ---------------------------------------------------------------------------
__global__ __launch_bounds__(256)
void dir_attn_kernel(const float* __restrict__ dep,
                     const float* __restrict__ head,
                     const float* __restrict__ rep,
                     const unsigned char* __restrict__ rep_mask,
                     float* __restrict__ attn_out,
                     int S, int D)
{
    const int d = threadIdx.x;
    const int i = blockIdx.x;
    const int b = blockIdx.y;

    const size_t base = (size_t)b * S * D;
    const float  h    = head[base + (size_t)i * D + d];

    float num = 0.0f, den = 0.0f;
    const float inv_scale = 1.0f / ATT_SCALE;

#pragma unroll 2
    for (int j = 0; j < i; ++j) {
        // Speculative prefetch a few rows ahead of the streaming j-loop
        // (stays inside the workspace allocation; L2-resident working set).
        __builtin_prefetch(dep + base + (size_t)(j + 4) * D + d, 0, 1);
        if (rep_mask[b * S + j]) {
            const float dj = dep[base + (size_t)j * D + d];
            const float rj = rep[base + (size_t)j * D + d];
            const float t  = ATT_SCALE * tanhf((dj + h) * inv_scale);
            const float e  = __expf(t);
            num = fmaf(e, rj, num);
            den += e;
        }
    }

    const float res = (den > 0.0f) ? (num / den) : 0.0f;
    attn_out[base + (size_t)i * D + d] = res;
}

// ---------------------------------------------------------------------------
// Fusion gate: one WMMA accumulator per N-tile takes both GEMMs
//   acc = attn * Wfa + rep * Wfr        (K doubled logically)
//   gate = sigmoid(acc + f_bias);  out = (gate*rep + (1-gate)*attn) * mask
// Same 16x32-per-wave structure: 2 A loads feed 4 WMMAs per k-step.
// ---------------------------------------------------------------------------
__global__ __launch_bounds__(128)
void fusion_wmma(const float* __restrict__ attn, const float* __restrict__ Wfa,
                 const float* __restrict__ rep,  const float* __restrict__ Wfr,
                 const float* __restrict__ fbias,
                 const unsigned char* __restrict__ rep_mask,
                 float* __restrict__ out,
                 int M, int K, int N)
{
    const int lane = threadIdx.x & 31;
    const int wave = threadIdx.x >> 5;
    const int lm   = lane & 15;
    const int half = lane >> 4;

    const int m0 = blockIdx.y * 16;
    const int n0 = (blockIdx.x * 4 + wave) * 32;

    const float* __restrict__ A1 = attn + (size_t)(m0 + lm) * K + 2 * half;
    const float* __restrict__ A2 = rep  + (size_t)(m0 + lm) * K + 2 * half;
    const float* __restrict__ Wa0 = Wfa + n0 + lm;
    const float* __restrict__ Wa1 = Wfa + n0 + 16 + lm;
    const float* __restrict__ Wr0 = Wfr + n0 + lm;
    const float* __restrict__ Wr1 = Wfr + n0 + 16 + lm;

    v8f c0 = {}, c1 = {};
#pragma unroll 2
    for (int k = 0; k < K; k += 4) {
        const int kk = k + 2 * half;

        v2f a1 = *(const v2f*)(A1 + k);
        v2f b0, b1;
        b0.x = Wa0[(size_t)kk * N];
        b0.y = Wa0[(size_t)(kk + 1) * N];
        b1.x = Wa1[(size_t)kk * N];
        b1.y = Wa1[(size_t)(kk + 1) * N];
        c0 = __builtin_amdgcn_wmma_f32_16x16x4_f32(false, a1, false, b0,
                                                   (short)0, c0, false, false);
        c1 = __builtin_amdgcn_wmma_f32_16x16x4_f32(false, a1, false, b1,
                                                   (short)0, c1, false, false);

        v2f a2 = *(const v2f*)(A2 + k);
        b0.x = Wr0[(size_t)kk * N];
        b0.y = Wr0[(size_t)(kk + 1) * N];
        b1.x = Wr1[(size_t)kk * N];
        b1.y = Wr1[(size_t)(kk + 1) * N];
        c0 = __builtin_amdgcn_wmma_f32_16x16x4_f32(false, a2, false, b0,
                                                   (short)0, c0, false, false);
        c1 = __builtin_amdgcn_wmma_f32_16x16x4_f32(false, a2, false, b1,
                                                   (short)0, c1, false, false);
    }

    const int col0 = n0 + lm;
    const int col1 = n0 + 16 + lm;
    const float bv0 = fbias[col0];
    const float bv1 = fbias[col1];

#pragma unroll
    for (int r = 0; r < 8; ++r) {
        const int row = m0 + 8 * half + r;          // row == b*S + s; mask is [B*S]
        const float mk = (float)rep_mask[row];

        const float g0  = 1.0f / (1.0f + __expf(-(c0[r] + bv0)));
        const float rm0 = rep[(size_t)row * N + col0];
        const float av0 = attn[(size_t)row * N + col0];
        out[(size_t)row * N + col0] = (g0 * rm0 + (1.0f - g0) * av0) * mk;

        const float g1  = 1.0f / (1.0f + __expf(-(c1[r] + bv1)));
        const float rm1 = rep[(size_t)row * N + col1];
        const float av1 = attn[(size_t)row * N + col1];
        out[(size_t)row * N + col1] = (g1 * rm1 + (1.0f - g1) * av1) * mk;
    }
}

// ---------------------------------------------------------------------------
// Launch
// ---------------------------------------------------------------------------
extern "C" void kernel_launch(void* const* d_in, const int* in_sizes, int n_in,
                              void* d_out, int out_size, void* d_ws, size_t ws_size,
                              hipStream_t stream)
{
    (void)in_sizes; (void)n_in; (void)out_size; (void)ws_size;

    constexpr int B = 16, S = 128, D = 256;
    constexpr int M = B * S;                        // 2048

    const float*         x        = (const float*)d_in[0];
    const unsigned char* rep_mask = (const unsigned char*)d_in[1];  // jnp bool_ = 1 byte
    const float*         Wr       = (const float*)d_in[2];  // rep_map_kernel
    const float*         br       = (const float*)d_in[3];  // rep_map_bias
    const float*         Wh       = (const float*)d_in[4];  // head_kernel
    const float*         Wd       = (const float*)d_in[5];  // dependent_kernel
    const float*         attn_b   = (const float*)d_in[6];  // attn_bias
    const float*         Wfr      = (const float*)d_in[7];  // f_rep_kernel
    const float*         Wfa      = (const float*)d_in[8];  // f_attn_kernel
    const float*         fb       = (const float*)d_in[9];  // f_bias
    float*               out      = (float*)d_out;

    float* rep  = (float*)d_ws;                     // [M,D]
    float* dep  = rep  + (size_t)M * D;             // [M,D] (includes attn_bias)
    float* head = dep  + (size_t)M * D;             // [M,D]
    float* attn = head + (size_t)M * D;             // [M,D]

    const dim3 blk(128);
    const dim3 grd(D / 128, M / 16);                // 2 x 128 = 256 blocks, 4 waves each

    // rep_map = elu(x @ Wr + br)
    gemm16x32_wmma<EPI_ELU_BIAS><<<grd, blk, 0, stream>>>(x, Wr, br, rep, M, D, D);
    // dependent = rep_map @ Wd + attn_bias   (bias folded here)
    gemm16x32_wmma<EPI_BIAS><<<grd, blk, 0, stream>>>(rep, Wd, attn_b, dep, M, D, D);
    // head = rep_map @ Wh
    gemm16x32_wmma<EPI_NONE><<<grd, blk, 0, stream>>>(rep, Wh, nullptr, head, M, D, D);

    // fused directional tanh-softmax attention (VALU/transcendental bound)
    dir_attn_kernel<<<dim3(S, B), dim3(D), 0, stream>>>(dep, head, rep, rep_mask,
                                                        attn, S, D);

    // fusion gate + blend + output mask
    fusion_wmma<<<grd, blk, 0, stream>>>(attn, Wfa, rep, Wfr, fb, rep_mask,
                                         out, M, D, D);
}